// KHopCoordExtractor_38972533244288
// MI455X (gfx1250) — compile-verified
//
#include <hip/hip_runtime.h>
#include <math.h>

typedef float v2f __attribute__((ext_vector_type(2)));
typedef float v8f __attribute__((ext_vector_type(8)));

#define MM 768
#define NN 2048
#define DD 128

__device__ __forceinline__ float silu_f(float x) {
    return x / (1.0f + __expf(-x));
}

// ---------------------------------------------------------------------------
// Gather: h[m] = x[idx[m]], c[m] = coords[idx[m]]
__global__ void gather_kernel(const float* __restrict__ x,
                              const float* __restrict__ coords,
                              const int* __restrict__ idx,
                              float* __restrict__ h, float* __restrict__ c) {
    int m = blockIdx.x;
    int k = threadIdx.x;            // DD threads
    int s = idx[m];
    h[m * DD + k] = x[(size_t)s * DD + k];
    if (k < 3) c[m * 3 + k] = coords[(size_t)s * 3 + k];
}

// ---------------------------------------------------------------------------
// Pairwise safe-norm distances: dist[i,j]
__global__ void dist_kernel(const float* __restrict__ c, float* __restrict__ dist) {
    int j = blockIdx.x * blockDim.x + threadIdx.x;
    int i = blockIdx.y;
    if (j < MM) {
        float dx = c[i * 3 + 0] - c[j * 3 + 0];
        float dy = c[i * 3 + 1] - c[j * 3 + 1];
        float dz = c[i * 3 + 2] - c[j * 3 + 2];
        float sq = dx * dx + dy * dy + dz * dz;
        dist[(size_t)i * MM + j] = (sq > 0.0f) ? sqrtf(sq) : 0.0f;
    }
}

// ---------------------------------------------------------------------------
// fp32 WMMA GEMM: Dst[M,N] = A[M,K] @ B[K,N] (+ bias[col]) (+ addsrc elementwise)
// One wave per 16x16 tile; block = 128 threads (4 waves -> 4 consecutive M-tiles).
// Uses V_WMMA_F32_16X16X4_F32 (exact fp32).
__global__ void wmma_gemm_f32(const float* __restrict__ A,
                              const float* __restrict__ B,
                              const float* __restrict__ bias,
                              float* __restrict__ Dst,
                              const float* __restrict__ addsrc,
                              int K, int N) {
    int wave   = threadIdx.x >> 5;
    int lane   = threadIdx.x & 31;
    int half   = lane >> 4;          // 0: lanes 0-15, 1: lanes 16-31
    int r      = lane & 15;
    int tile_n = blockIdx.x;
    int tile_m = blockIdx.y * 4 + wave;
    int m0 = tile_m * 16;
    int n0 = tile_n * 16;

    v8f acc = {};
    for (int k0 = 0; k0 < K; k0 += 4) {
        // A 16x4 tile: VGPR0 holds K = 2*half, VGPR1 holds K = 2*half+1, row = m0+r
        const float* Ap = A + (size_t)(m0 + r) * K + (k0 + 2 * half);
        v2f a; a.x = Ap[0]; a.y = Ap[1];
        // B 4x16 tile (row-major, stride N): mirrored K split across lane halves
        const float* Bp = B + (size_t)(k0 + 2 * half) * N + (n0 + r);
        v2f b; b.x = Bp[0]; b.y = Bp[N];
        acc = __builtin_amdgcn_wmma_f32_16x16x4_f32(
            false, a, false, b, (short)0, acc, false, false);
    }
    // D layout: VGPR v -> row m0 + half*8 + v, col n0 + r
#pragma unroll
    for (int v = 0; v < 8; ++v) {
        int row = m0 + half * 8 + v;
        int col = n0 + r;
        float val = acc[v];
        if (bias)   val += bias[col];
        if (addsrc) val += addsrc[(size_t)row * N + col];
        Dst[(size_t)row * N + col] = val;
    }
}

// ---------------------------------------------------------------------------
// phi_c scalar projections: cax[i] = h[i,:]·Wc[0:D], cbx[i] = h[i,:]·Wc[D:2D]
__global__ void phi_c_proj(const float* __restrict__ h, const float* __restrict__ Wc,
                           float* __restrict__ cax, float* __restrict__ cbx) {
    __shared__ float s1[DD], s2[DD];
    int i = blockIdx.x;
    int k = threadIdx.x;            // DD threads
    float hv = h[(size_t)i * DD + k];
    s1[k] = hv * Wc[k];
    s2[k] = hv * Wc[DD + k];
    __syncthreads();
    for (int s = DD / 2; s > 0; s >>= 1) {
        if (k < s) { s1[k] += s1[k + s]; s2[k] += s2[k + s]; }
        __syncthreads();
    }
    if (k == 0) { cax[i] = s1[0]; cbx[i] = s2[0]; }
}

// ---------------------------------------------------------------------------
// Fused message reduction (VALU/transcendental-bound hot loop):
// hidmean[i,k] = (1/M) * sum_j silu(ax[i,k] + bx[j,k] + dist[i,j]*w1c[k])
// Block = node i (128 threads = 4 waves). Each thread owns 4 consecutive
// channels (float4 / global_load_b128); wave w handles j strip (j % 4 == w).
// Cross-wave partials reduced through LDS.
__global__ void msg_kernel(const float* __restrict__ ax, const float* __restrict__ bx,
                           const float* __restrict__ dist, const float* __restrict__ w1c,
                           float* __restrict__ hidmean) {
    __shared__ float red[4][DD];
    int i    = blockIdx.x;
    int w    = threadIdx.x >> 5;    // wave id 0..3 -> j strip
    int lane = threadIdx.x & 31;
    int k4   = lane * 4;

    float4 ai = ((const float4*)(ax + (size_t)i * DD))[lane];
    float4 wk = ((const float4*)w1c)[lane];
    const float* di = dist + (size_t)i * MM;

    float4 acc = make_float4(0.0f, 0.0f, 0.0f, 0.0f);
    for (int j = w; j < MM; j += 4) {
        if (j + 32 < MM) __builtin_prefetch(bx + (size_t)(j + 32) * DD + k4, 0, 1);
        float  d = di[j];
        float4 b = ((const float4*)(bx + (size_t)j * DD))[lane];
        acc.x += silu_f(ai.x + b.x + d * wk.x);
        acc.y += silu_f(ai.y + b.y + d * wk.y);
        acc.z += silu_f(ai.z + b.z + d * wk.z);
        acc.w += silu_f(ai.w + b.w + d * wk.w);
    }
    red[w][k4 + 0] = acc.x;
    red[w][k4 + 1] = acc.y;
    red[w][k4 + 2] = acc.z;
    red[w][k4 + 3] = acc.w;
    __syncthreads();
    int t = threadIdx.x;            // DD == blockDim.x
    float s = red[0][t] + red[1][t] + red[2][t] + red[3][t];
    hidmean[(size_t)i * DD + t] = s * (1.0f / (float)MM);
}

// ---------------------------------------------------------------------------
// Coordinate update: c_next[i] = c[i] + (1/M) * sum_j silu(...) * (c[i]-c[j])
__global__ void coord_kernel(const float* __restrict__ c,
                             const float* __restrict__ cax, const float* __restrict__ cbx,
                             const float* __restrict__ dist,
                             const float* __restrict__ wccp, const float* __restrict__ cbp,
                             float* __restrict__ c_next) {
    __shared__ float sx[256], sy[256], sz[256];
    int i = blockIdx.x;
    int t = threadIdx.x;            // 256 threads
    float wcc = *wccp;
    float cb0 = *cbp;
    float cix = c[i * 3 + 0], ciy = c[i * 3 + 1], ciz = c[i * 3 + 2];
    float cai = cax[i];
    float axx = 0.0f, ayy = 0.0f, azz = 0.0f;
    for (int j = t; j < MM; j += 256) {
        float d   = dist[(size_t)i * MM + j];
        float w   = silu_f(cai + cbx[j] + d * wcc + cb0);
        axx += w * (cix - c[j * 3 + 0]);
        ayy += w * (ciy - c[j * 3 + 1]);
        azz += w * (ciz - c[j * 3 + 2]);
    }
    sx[t] = axx; sy[t] = ayy; sz[t] = azz;
    __syncthreads();
    for (int s = 128; s > 0; s >>= 1) {
        if (t < s) { sx[t] += sx[t + s]; sy[t] += sy[t + s]; sz[t] += sz[t + s]; }
        __syncthreads();
    }
    if (t == 0) {
        float inv = 1.0f / (float)MM;
        c_next[i * 3 + 0] = cix + sx[0] * inv;
        c_next[i * 3 + 1] = ciy + sy[0] * inv;
        c_next[i * 3 + 2] = ciz + sz[0] * inv;
    }
}

// ---------------------------------------------------------------------------
__global__ void zero_kernel(float* __restrict__ p, int n) {
    int i = blockIdx.x * blockDim.x + threadIdx.x;
    if (i < n) p[i] = 0.0f;
}

__global__ void scatter_kernel(const float* __restrict__ h, const int* __restrict__ seg,
                               float* __restrict__ sums, float* __restrict__ cnt) {
    int m = blockIdx.x;
    int k = threadIdx.x;            // DD threads
    int s = seg[m];
    atomicAdd(&sums[(size_t)s * DD + k], h[(size_t)m * DD + k]);
    if (k == 0) atomicAdd(&cnt[s], 1.0f);
}

__global__ void pool_kernel(const float* __restrict__ sums, const float* __restrict__ cnt,
                            float* __restrict__ hpool) {
    int n = blockIdx.x;
    int k = threadIdx.x;            // DD threads
    float ct = cnt[n];
    hpool[(size_t)n * DD + k] = (ct > 0.0f) ? sums[(size_t)n * DD + k] / ct : 0.0f;
}

// BatchNorm batch stats per column of concat([x, hpool]); emit scale/shift.
__global__ void bn_stats(const float* __restrict__ x, const float* __restrict__ hpool,
                         const float* __restrict__ gamma, const float* __restrict__ beta,
                         float* __restrict__ scale, float* __restrict__ shift) {
    __shared__ float s1[256], s2[256];
    int c = blockIdx.x;             // 2*DD columns
    int t = threadIdx.x;            // 256 threads
    float sum = 0.0f, sq = 0.0f;
    for (int n = t; n < NN; n += 256) {
        float v = (c < DD) ? x[(size_t)n * DD + c] : hpool[(size_t)n * DD + (c - DD)];
        sum += v; sq += v * v;
    }
    s1[t] = sum; s2[t] = sq;
    __syncthreads();
    for (int s = 128; s > 0; s >>= 1) {
        if (t < s) { s1[t] += s1[t + s]; s2[t] += s2[t + s]; }
        __syncthreads();
    }
    if (t == 0) {
        float mu  = s1[0] / (float)NN;
        float var = s2[0] / (float)NN - mu * mu;
        float sc  = gamma[c] / sqrtf(var + 1e-5f);
        scale[c] = sc;
        shift[c] = beta[c] - mu * sc;
    }
}

// Build normalized concat matrix [NN, 2*DD]
__global__ void norm_kernel(const float* __restrict__ x, const float* __restrict__ hpool,
                            const float* __restrict__ scale, const float* __restrict__ shift,
                            float* __restrict__ normed) {
    int n = blockIdx.x;
    int t = threadIdx.x;            // 2*DD threads
    float v = (t < DD) ? x[(size_t)n * DD + t] : hpool[(size_t)n * DD + (t - DD)];
    normed[(size_t)n * (2 * DD) + t] = v * scale[t] + shift[t];
}

// ---------------------------------------------------------------------------
extern "C" void kernel_launch(void* const* d_in, const int* in_sizes, int n_in,
                              void* d_out, int out_size, void* d_ws, size_t ws_size,
                              hipStream_t stream) {
    const float* x      = (const float*)d_in[0];
    const float* coords = (const float*)d_in[1];
    const float* w1     = (const float*)d_in[2];   // [L, 2D+1, D]
    const float* b1     = (const float*)d_in[3];   // [L, D]
    const float* w2     = (const float*)d_in[4];   // [L, D, D]
    const float* b2     = (const float*)d_in[5];   // [L, D]
    const float* cw     = (const float*)d_in[6];   // [L, 2D+1, 1]
    const float* cb     = (const float*)d_in[7];   // [L, 1]
    const float* gamma  = (const float*)d_in[8];   // [2D]
    const float* beta   = (const float*)d_in[9];   // [2D]
    const float* outw   = (const float*)d_in[10];  // [2D, D]
    const float* outb   = (const float*)d_in[11];  // [D]
    const int*   nidx   = (const int*)d_in[12];    // [M]
    const int*   sidx   = (const int*)d_in[13];    // [M]
    (void)in_sizes; (void)n_in; (void)out_size; (void)ws_size;

    const int L = 3;

    char* wsp = (char*)d_ws;
    auto alloc = [&](size_t bytes) -> float* {
        char* p = wsp;
        wsp += (bytes + 255) & ~(size_t)255;
        return (float*)p;
    };
    float* h     = alloc((size_t)MM * DD * 4);
    float* dist  = alloc((size_t)MM * MM * 4);
    float* c0    = alloc((size_t)MM * 3 * 4);
    float* c1    = alloc((size_t)MM * 3 * 4);
    float* ax    = alloc((size_t)MM * DD * 4);
    float* bx    = alloc((size_t)MM * DD * 4);
    float* hm    = alloc((size_t)MM * DD * 4);
    float* cax   = alloc((size_t)MM * 4);
    float* cbx   = alloc((size_t)MM * 4);
    float* sums  = alloc((size_t)NN * DD * 4);   // cnt placed right after (both 256B-mult)
    float* cnt   = alloc((size_t)NN * 4);
    float* hpool = alloc((size_t)NN * DD * 4);
    float* scale = alloc((size_t)2 * DD * 4);
    float* shift = alloc((size_t)2 * DD * 4);
    float* nrm   = alloc((size_t)NN * 2 * DD * 4);

    gather_kernel<<<MM, DD, 0, stream>>>(x, coords, nidx, h, c0);

    float* ccur = c0;
    float* cnxt = c1;
    for (int l = 0; l < L; ++l) {
        const float* w1l = w1 + (size_t)l * (2 * DD + 1) * DD;
        const float* cwl = cw + (size_t)l * (2 * DD + 1);

        dist_kernel<<<dim3(MM / 256, MM), 256, 0, stream>>>(ccur, dist);
        // ax = h @ W1[:D] + b1   (768x128 @ 128x128, fp32 WMMA)
        wmma_gemm_f32<<<dim3(DD / 16, MM / 64), 128, 0, stream>>>(
            h, w1l, b1 + (size_t)l * DD, ax, nullptr, DD, DD);
        // bx = h @ W1[D:2D]
        wmma_gemm_f32<<<dim3(DD / 16, MM / 64), 128, 0, stream>>>(
            h, w1l + (size_t)DD * DD, nullptr, bx, nullptr, DD, DD);
        phi_c_proj<<<MM, DD, 0, stream>>>(h, cwl, cax, cbx);
        // fused silu + mean over j
        msg_kernel<<<MM, DD, 0, stream>>>(ax, bx, dist, w1l + (size_t)2 * DD * DD, hm);
        // coord gate + update (writes double-buffered c)
        coord_kernel<<<MM, 256, 0, stream>>>(ccur, cax, cbx, dist,
                                             cwl + 2 * DD, cb + l, cnxt);
        // h = h + hm @ W2 + b2   (fp32 WMMA with residual add)
        wmma_gemm_f32<<<dim3(DD / 16, MM / 64), 128, 0, stream>>>(
            hm, w2 + (size_t)l * DD * DD, b2 + (size_t)l * DD, h, h, DD, DD);
        float* t = ccur; ccur = cnxt; cnxt = t;
    }

    // scatter-mean over indicator index
    int zn = NN * DD + NN;   // sums and cnt are contiguous
    zero_kernel<<<(zn + 255) / 256, 256, 0, stream>>>(sums, zn);
    scatter_kernel<<<MM, DD, 0, stream>>>(h, sidx, sums, cnt);
    pool_kernel<<<NN, DD, 0, stream>>>(sums, cnt, hpool);

    // BatchNorm (training stats) fused into scale/shift, then final projection
    bn_stats<<<2 * DD, 256, 0, stream>>>(x, hpool, gamma, beta, scale, shift);
    norm_kernel<<<NN, 2 * DD, 0, stream>>>(x, hpool, scale, shift, nrm);
    // out = normed[2048,256] @ out_w[256,128] + out_b  (fp32 WMMA)
    wmma_gemm_f32<<<dim3(DD / 16, NN / 64), 128, 0, stream>>>(
        nrm, outw, outb, (float*)d_out, nullptr, 2 * DD, DD);
}